// SparseMultiheadAttention_33758442946704
// MI455X (gfx1250) — compile-verified
//
#include <hip/hip_runtime.h>
#include <math.h>

// ---------------------------------------------------------------------------
// SparseMultiheadAttention (local banded attention, SPAN=50) for gfx1250.
// Full-f32 pipeline using V_WMMA_F32_16X16X4_F32 (wave32 WMMA).
// GEMMs stage BOTH operands into LDS with the Tensor Data Mover (two parallel
// per-wave DMAs, TENSORcnt-tracked), using the TDM LDS-padding feature to get
// bank-conflict-friendly pitches. Inner K-loop is pure LDS -> WMMA.
// ---------------------------------------------------------------------------

typedef __attribute__((ext_vector_type(2))) float v2f;
typedef __attribute__((ext_vector_type(8))) float v8f;
typedef __attribute__((ext_vector_type(4))) unsigned int u32x4;
typedef __attribute__((ext_vector_type(8))) int i32x8;
typedef __attribute__((ext_vector_type(4))) int i32x4;

#define SPAN 50
#define EDIM 128
#define HEADS 8
#define DHEAD 16
#define WPITCH 144  // W panel LDS pitch (DWORDs): 128 + 16 pad -> halves 32 banks apart

static __device__ __forceinline__ v8f wmma_f32(v2f a, v2f b, v8f c) {
  // 8 args: (neg_a, A, neg_b, B, c_mod, C, reuse_a, reuse_b)
  return __builtin_amdgcn_wmma_f32_16x16x4_f32(false, a, false, b, (short)0, c,
                                               false, false);
}

#if __has_builtin(__builtin_amdgcn_tensor_load_to_lds)
#define HAVE_TDM 1
// 2-D TDM tile load: rows x cols f32 tile at gptr (row stride row_stride
// elems) -> LDS byte offset lds_byte_off, optionally padding the LDS
// destination by (pad_amt_code+1) DWORDs every 2^(pad_int_code+1) DWORDs.
// Descriptor layout per CDNA5 ISA (D# groups 0/1).
// This toolchain exposes the 6-arg builtin form.
static __device__ __forceinline__ void tdm_load_tile_f32(
    const float* gptr, unsigned lds_byte_off, unsigned rows, unsigned cols,
    unsigned row_stride, unsigned pad_en, unsigned pad_int_code,
    unsigned pad_amt_code) {
  const unsigned long long ga = (unsigned long long)(uintptr_t)gptr;
  u32x4 g0;
  g0.x = 1u;                                  // count=1 valid descriptor
  g0.y = lds_byte_off;                        // lds_addr (bytes)
  g0.z = (unsigned)(ga & 0xFFFFFFFFu);        // global_addr[31:0]
  g0.w = (unsigned)((ga >> 32) & 0x01FFFFFFu) // global_addr[56:32]
         | (2u << 30);                        // type=2 ("image")
  i32x8 g1;
  g1[0] = (int)((2u << 16) |                  // wg_mask=0, data_size=2 (4B)
                (pad_en << 20) |              // pad_enable
                (pad_int_code << 22) |        // pad_interval
                (pad_amt_code << 25));        // pad_amount
  g1[1] = (int)((cols & 0xFFFFu) << 16);      // tensor_dim0[15:0]
  g1[2] = (int)(((cols >> 16) & 0xFFFFu) | ((rows & 0xFFFFu) << 16));
  g1[3] = (int)(((rows >> 16) & 0xFFFFu) | ((cols & 0xFFFFu) << 16)); // tile_dim0
  g1[4] = (int)(rows & 0xFFFFu);              // tile_dim1 (tile_dim2=0)
  g1[5] = (int)row_stride;                    // tensor_dim0_stride[31:0]
  g1[6] = 0;                                  // stride hi / dim1_stride
  g1[7] = 0;
  const i32x4 gz4 = {0, 0, 0, 0};             // 2-D tensor: groups 2/3 unused
  const i32x8 gz8 = {0, 0, 0, 0, 0, 0, 0, 0};
  __builtin_amdgcn_tensor_load_to_lds(g0, g1, gz4, gz4, gz8, 0);
}
#else
#define HAVE_TDM 0
#endif

// ---------------------------------------------------------------------------
// Y[n x E] = X[n x K] @ W[K x E] + bias[E]     (E == 128, K in {128, 8})
// 256 threads = 8 waves; wave w owns the (blockIdx.x, w) 16x16 tile.
// LDS: s_x = 16 x (K+2) tile of X     (TDM by wave 0, 2-DWORD row pad)
//      s_w = K  x WPITCH panel of W   (TDM by wave 1, 16-DWORD row pad)
// Inner loop: A frag = ds_load_b64 from s_x, B frag = 2x ds_load from s_w.
// ---------------------------------------------------------------------------
__global__ __launch_bounds__(256) void gemm_bias_f32_wmma(
    const float* __restrict__ X, const float* __restrict__ W,
    const float* __restrict__ bias, float* __restrict__ Y, int n, int K) {
  extern __shared__ float s_lds[];

  const int lane = threadIdx.x & 31;
  const int wave = threadIdx.x >> 5;
  const int r16  = lane & 15;
  const int half = lane >> 4;
  const int m0   = blockIdx.x * 16;
  const int n0   = wave * 16;  // E/16 == 8 waves

  const int xpitch = K + 2;                 // padded X pitch (DWORDs)
  float* s_x = s_lds;                       // 16 * xpitch floats
  float* s_w = s_lds + 16 * xpitch;         // K * WPITCH floats

#if HAVE_TDM
  if (wave == 0) {
    // X tile: rows of K DWORDs, pad 2 DWORDs each row -> pitch K+2.
    // pad_interval code: 2^(c+1) DWORDs == K  ->  c = ctz(K)-1 (128->6, 8->2).
    tdm_load_tile_f32(X + (size_t)m0 * K, 0u, 16u, (unsigned)K, (unsigned)K,
                      1u, (unsigned)(__builtin_ctz((unsigned)K) - 1), 1u);
    __builtin_amdgcn_s_wait_tensorcnt(0);
  } else if (wave == 1) {
    // Full K x 128 weight panel, pad 16 DWORDs every 128 -> pitch WPITCH.
    tdm_load_tile_f32(W, (unsigned)(16 * xpitch * sizeof(float)), (unsigned)K,
                      128u, 128u, 1u, 6u, 15u);
    __builtin_amdgcn_s_wait_tensorcnt(0);
  }
#else
  for (int idx = threadIdx.x; idx < 16 * K; idx += 256) {
    s_x[(idx / K) * xpitch + (idx % K)] = X[(size_t)m0 * K + idx];
  }
  for (int idx = threadIdx.x; idx < K * 128; idx += 256) {
    s_w[(idx >> 7) * WPITCH + (idx & 127)] = W[idx];
  }
#endif
  __syncthreads();

  v8f acc = {};
  for (int k0 = 0; k0 < K; k0 += 4) {
    const int kb = k0 + 2 * half;
    const v2f a = *(const v2f*)&s_x[r16 * xpitch + kb];  // ds_load_b64
    v2f b;
    b.x = s_w[kb * WPITCH + n0 + r16];
    b.y = s_w[(kb + 1) * WPITCH + n0 + r16];
    acc = wmma_f32(a, b, acc);
  }

  const float bv = bias[n0 + r16];
#pragma unroll
  for (int i = 0; i < 8; ++i) {
    const int row = m0 + i + 8 * half;
    Y[(size_t)row * EDIM + n0 + r16] = acc[i] + bv;
  }
}

// ---------------------------------------------------------------------------
// Local banded attention for one (16-query tile, head) per block.
// Window union of rows i0..i0+15 spans keys [i0-50, i0+65] -> 116, pad to 128.
// scores = (Q @ Kwin^T) / sqrt(n) + (B @ Bwin^T), banded mask, softmax,
// out = P @ Vwin.  128 threads = 4 waves.
// ---------------------------------------------------------------------------
__global__ __launch_bounds__(128) void local_attn_f32_wmma(
    const float* __restrict__ qp, const float* __restrict__ kp,
    const float* __restrict__ vp, const float* __restrict__ bp,
    float* __restrict__ ap, int n) {
  __shared__ float s_scores[16][128];
  __shared__ float s_out[16][16];

  const int tid  = threadIdx.x;
  const int lane = tid & 31;
  const int wave = tid >> 5;
  const int r16  = lane & 15;
  const int half = lane >> 4;
  const int i0   = blockIdx.x * 16;
  const int h    = blockIdx.y;
  const int hc   = h * DHEAD;

  // zero the cross-wave output accumulator (256 entries, 128 threads)
  ((float*)s_out)[tid]       = 0.0f;
  ((float*)s_out)[tid + 128] = 0.0f;

  const float scale  = rsqrtf((float)n);  // reference scales q.k by 1/sqrt(n)
  const float NEGINF = -__builtin_inff();

  // ---- scores: each wave computes two 16-key N-tiles --------------------
  const float* __restrict__ qrow = qp + (size_t)(i0 + r16) * EDIM + hc;
  const float* __restrict__ brow = bp + (size_t)(i0 + r16) * EDIM + hc;

  for (int t = wave * 2; t < wave * 2 + 2; ++t) {
    const int c0 = t * 16;
    const int j  = i0 - SPAN + c0 + r16;          // key index of this B column
    const int jc = min(max(j, 0), n - 1);         // clamp like reference cols_c
    const float* __restrict__ kcol = kp + (size_t)jc * EDIM + hc;
    const float* __restrict__ bcol = bp + (size_t)jc * EDIM + hc;

    v8f sqk = {}, sbb = {};
#pragma unroll
    for (int k0 = 0; k0 < DHEAD; k0 += 4) {
      const int kb = k0 + 2 * half;
      v2f a, b;
      a.x = qrow[kb];
      a.y = qrow[kb + 1];
      b.x = kcol[kb];
      b.y = kcol[kb + 1];
      sqk = wmma_f32(a, b, sqk);
      a.x = brow[kb];
      a.y = brow[kb + 1];
      b.x = bcol[kb];
      b.y = bcol[kb + 1];
      sbb = wmma_f32(a, b, sbb);
    }
#pragma unroll
    for (int i = 0; i < 8; ++i) {
      const int row = i + 8 * half;   // query row within tile
      const int col = c0 + r16;       // window column 0..127
      const int jj  = i0 - SPAN + col;
      const bool valid =
          (col >= row) && (col <= row + 2 * SPAN) && (jj >= 0) && (jj < n);
      const float val = sqk[i] * scale + sbb[i];
      s_scores[row][col] = valid ? val : NEGINF;
    }
  }
  __syncthreads();

  // ---- row softmax over the 128-wide padded window ----------------------
  if (tid < 16) {
    float m = NEGINF;
    for (int c = 0; c < 128; ++c) m = fmaxf(m, s_scores[tid][c]);
    float sum = 0.0f;
    for (int c = 0; c < 128; ++c) {
      const float e = __expf(s_scores[tid][c] - m);  // exp(-inf) -> 0
      s_scores[tid][c] = e;
      sum += e;
    }
    const float inv = 1.0f / sum;
    for (int c = 0; c < 128; ++c) s_scores[tid][c] *= inv;
  }
  __syncthreads();

  // ---- out = P (16x128) @ Vwin (128x16); 32 K-positions per wave --------
  v8f o = {};
  for (int kk = wave * 32; kk < wave * 32 + 32; kk += 4) {
    const int kb = kk + 2 * half;
    v2f a, b;
    a.x = s_scores[r16][kb];
    a.y = s_scores[r16][kb + 1];
    const int j0  = i0 - SPAN + kb;
    const int jc0 = min(max(j0, 0), n - 1);
    const int jc1 = min(max(j0 + 1, 0), n - 1);
    b.x = vp[(size_t)jc0 * EDIM + hc + r16];
    b.y = vp[(size_t)jc1 * EDIM + hc + r16];
    o = wmma_f32(a, b, o);
  }
#pragma unroll
  for (int i = 0; i < 8; ++i) {
    atomicAdd(&s_out[i + 8 * half][r16], o[i]);
  }
  __syncthreads();

  // ---- write the head tile to the pre-Wo activation buffer --------------
  {
    int row = tid >> 4, col = tid & 15;
    ap[(size_t)(i0 + row) * EDIM + hc + col] = s_out[row][col];
    row = (tid + 128) >> 4;
    ap[(size_t)(i0 + row) * EDIM + hc + col] = s_out[row][col];
  }
}

// ---------------------------------------------------------------------------
extern "C" void kernel_launch(void* const* d_in, const int* in_sizes, int n_in,
                              void* d_out, int out_size, void* d_ws,
                              size_t ws_size, hipStream_t stream) {
  const float* query = (const float*)d_in[0];
  const float* key   = (const float*)d_in[1];
  const float* value = (const float*)d_in[2];
  const float* abias = (const float*)d_in[3];
  const float* Wq = (const float*)d_in[4];
  const float* bq = (const float*)d_in[5];
  const float* Wk = (const float*)d_in[6];
  const float* bk = (const float*)d_in[7];
  const float* Wv = (const float*)d_in[8];
  const float* bv = (const float*)d_in[9];
  const float* Wo = (const float*)d_in[10];
  const float* bo = (const float*)d_in[11];
  const float* Wfe = (const float*)d_in[12];
  const float* bfe = (const float*)d_in[13];
  float* out = (float*)d_out;

  const int n = in_sizes[0] / EDIM;  // 4096

  float* qp = (float*)d_ws;
  float* kp = qp + (size_t)n * EDIM;
  float* vp = kp + (size_t)n * EDIM;
  float* bp = vp + (size_t)n * EDIM;
  float* ap = bp + (size_t)n * EDIM;

  const dim3 gblk(256);
  const dim3 ggrd(n / 16);
  // dynamic LDS: 16*(K+2) X-tile + K*WPITCH W-panel (floats)
  const size_t ldsE = (16 * (EDIM + 2) + EDIM * WPITCH) * sizeof(float);
  const size_t ldsH = (16 * (HEADS + 2) + HEADS * WPITCH) * sizeof(float);

  gemm_bias_f32_wmma<<<ggrd, gblk, ldsE, stream>>>(query, Wq, bq, qp, n, EDIM);
  gemm_bias_f32_wmma<<<ggrd, gblk, ldsE, stream>>>(key, Wk, bk, kp, n, EDIM);
  gemm_bias_f32_wmma<<<ggrd, gblk, ldsE, stream>>>(value, Wv, bv, vp, n, EDIM);
  gemm_bias_f32_wmma<<<ggrd, gblk, ldsH, stream>>>(abias, Wfe, bfe, bp, n, HEADS);

  local_attn_f32_wmma<<<dim3(n / 16, HEADS), dim3(128), 0, stream>>>(
      qp, kp, vp, bp, ap, n);

  gemm_bias_f32_wmma<<<ggrd, gblk, ldsE, stream>>>(ap, Wo, bo, out, n, EDIM);
}